// GDSSM_8461085573502
// MI455X (gfx1250) — compile-verified
//
#include <hip/hip_runtime.h>

#define Nn 16384
#define Dd 300
#define Rr 10
#define Bb 1024
#define Kk 64

typedef __attribute__((ext_vector_type(2))) float v2f;
typedef __attribute__((ext_vector_type(8))) float v8f;

__device__ __forceinline__ float wave_sum32(float x) {
#pragma unroll
  for (int off = 1; off < 32; off <<= 1) x += __shfl_xor(x, off, 32);
  return x;
}

// ---------------------------------------------------------------------------
// Kernel A: Householder tower (10 reflections) + row L2-normalize.
// One wave per row; lane j handles elements d = lane + 32*j, j < 10.
// ---------------------------------------------------------------------------
__global__ __launch_bounds__(256) void tower_kernel(
    const float* __restrict__ feat_src, const float* __restrict__ feat_tgt,
    const float* __restrict__ vs_src, const float* __restrict__ vs_tgt,
    float* __restrict__ hn_src, float* __restrict__ hn_tgt) {
  const int wave = (blockIdx.x * blockDim.x + threadIdx.x) >> 5;
  const int lane = threadIdx.x & 31;
  const float* feat;
  const float* vs;
  float* hn;
  int row;
  if (wave < Nn) {
    feat = feat_src; vs = vs_src; hn = hn_src; row = wave;
  } else {
    feat = feat_tgt; vs = vs_tgt; hn = hn_tgt; row = wave - Nn;
  }
  const float* hrow = feat + (size_t)row * Dd;
  float h[10];
#pragma unroll
  for (int j = 0; j < 10; ++j) {
    int d = lane + 32 * j;
    h[j] = (d < Dd) ? hrow[d] : 0.0f;
  }
  for (int r = 0; r < Rr; ++r) {
    const float* v = vs + (size_t)r * Dd;
    float vj[10];
    float hv = 0.0f, vv = 0.0f;
#pragma unroll
    for (int j = 0; j < 10; ++j) {
      int d = lane + 32 * j;
      float x = (d < Dd) ? v[d] : 0.0f;
      vj[j] = x;
      hv += h[j] * x;
      vv += x * x;
    }
    hv = wave_sum32(hv);
    vv = wave_sum32(vv);
    float coef = hv / vv;
#pragma unroll
    for (int j = 0; j < 10; ++j) h[j] -= coef * vj[j];
  }
  float ss = 0.0f;
#pragma unroll
  for (int j = 0; j < 10; ++j) ss += h[j] * h[j];
  ss = wave_sum32(ss);
  float scale = 1.0f / fmaxf(sqrtf(ss), 1e-12f);
  float* orow = hn + (size_t)row * Dd;
#pragma unroll
  for (int j = 0; j < 10; ++j) {
    int d = lane + 32 * j;
    if (d < Dd) orow[d] = h[j] * scale;
  }
}

// ---------------------------------------------------------------------------
// Kernel B: fused GEMM (A @ B^T, fp32 WMMA 16x16x4) + per-row top-10 mean.
// Wave owns 16 rows, sweeps all 16384 columns in 16x64 blocks.
// Per-lane sorted top-10 lists (8 rows/lane per C-layout) in registers,
// merged across 16 lanes at the end via shuffles.
// ---------------------------------------------------------------------------
__device__ __forceinline__ void topk_insert(float (&lst)[10], float x) {
  if (x > lst[9]) {
#pragma unroll
    for (int i = 0; i < 10; ++i) {
      float mx = fmaxf(lst[i], x);
      float mn = fminf(lst[i], x);
      lst[i] = mx;
      x = mn;
    }
  }
}

__global__ __launch_bounds__(128) void csls_topk_kernel(
    const float* __restrict__ Am, const float* __restrict__ Bm,
    float* __restrict__ outr) {
  const int lane = threadIdx.x & 31;
  const int half = lane >> 4;   // 0: rows 0..7 of tile, 1: rows 8..15
  const int mrow = lane & 15;
  const int wave = (blockIdx.x * blockDim.x + threadIdx.x) >> 5;
  const int m0 = wave * 16;

  float top[8][10];
#pragma unroll
  for (int j = 0; j < 8; ++j)
#pragma unroll
    for (int i = 0; i < 10; ++i) top[j][i] = -3.0e38f;

  // A-fragment addressing: lane holds row m0+mrow, K elements {2*half, 2*half+1}
  const float* ap = Am + (size_t)(m0 + mrow) * Dd + 2 * half;

  for (int n0 = 0; n0 < Nn; n0 += 64) {
    const float* bp = Bm + (size_t)(n0 + mrow) * Dd + 2 * half;
    v8f c0 = {}, c1 = {}, c2 = {}, c3 = {};
    for (int k = 0; k < Dd; k += 4) {  // Dd = 300, divisible by 4
      v2f a  = *(const v2f*)(ap + k);
      v2f b0 = *(const v2f*)(bp + k);
      v2f b1 = *(const v2f*)(bp + 16 * Dd + k);
      v2f b2 = *(const v2f*)(bp + 32 * Dd + k);
      v2f b3 = *(const v2f*)(bp + 48 * Dd + k);
      c0 = __builtin_amdgcn_wmma_f32_16x16x4_f32(false, a, false, b0, (short)0, c0, false, false);
      c1 = __builtin_amdgcn_wmma_f32_16x16x4_f32(false, a, false, b1, (short)0, c1, false, false);
      c2 = __builtin_amdgcn_wmma_f32_16x16x4_f32(false, a, false, b2, (short)0, c2, false, false);
      c3 = __builtin_amdgcn_wmma_f32_16x16x4_f32(false, a, false, b3, (short)0, c3, false, false);
    }
    // C layout: lane (half h) VGPR j holds element (M = j + 8*h, N = n_tile + mrow)
#pragma unroll
    for (int j = 0; j < 8; ++j) {
      topk_insert(top[j], c0[j]);
      topk_insert(top[j], c1[j]);
      topk_insert(top[j], c2[j]);
      topk_insert(top[j], c3[j]);
    }
  }

  // Merge: for each of my 8 rows, the row's global top-10 lies in the union of
  // the 16 per-lane sorted lists of my half. 10 rounds of half-wide max.
#pragma unroll
  for (int j = 0; j < 8; ++j) {
    float sum = 0.0f;
    for (int it = 0; it < 10; ++it) {
      float head = top[j][0];
      float m = head;
      m = fmaxf(m, __shfl_xor(m, 1, 32));
      m = fmaxf(m, __shfl_xor(m, 2, 32));
      m = fmaxf(m, __shfl_xor(m, 4, 32));
      m = fmaxf(m, __shfl_xor(m, 8, 32));
      sum += m;
      unsigned long long ball = __ballot(head == m);
      unsigned bm = (unsigned)ball & (half ? 0xFFFF0000u : 0x0000FFFFu);
      int leader = __ffs(bm) - 1;
      if (lane == leader) {
        // pop my head (constant indices only; stays in registers)
#pragma unroll
        for (int i = 0; i < 9; ++i) top[j][i] = top[j][i + 1];
        top[j][9] = -3.0e38f;
      }
    }
    if (mrow == 0) outr[m0 + 8 * half + j] = sum * 0.1f;
  }
}

// ---------------------------------------------------------------------------
// Kernel C: gathered cosine sims + CSLS logits. One wave per (b,k) pair,
// computes both directions.
// ---------------------------------------------------------------------------
__global__ __launch_bounds__(256) void logits_kernel(
    const float* __restrict__ src_hn, const float* __restrict__ tgt_hn,
    const int* __restrict__ srcs_index, const int* __restrict__ tgts_index,
    const float* __restrict__ rt, const float* __restrict__ rs,
    float* __restrict__ out) {
  const int wave = (blockIdx.x * blockDim.x + threadIdx.x) >> 5;
  const int lane = threadIdx.x & 31;
  const int b = wave / Kk;
  const int k = wave % Kk;
  const int i_ps = srcs_index[b * Kk];       // pos src
  const int i_pt = tgts_index[b * Kk];       // pos tgt
  const int i_sk = srcs_index[b * Kk + k];   // src list
  const int i_tk = tgts_index[b * Kk + k];   // tgt list
  const float* ps = src_hn + (size_t)i_ps * Dd;
  const float* tk = tgt_hn + (size_t)i_tk * Dd;
  const float* pt = tgt_hn + (size_t)i_pt * Dd;
  const float* sk = src_hn + (size_t)i_sk * Dd;
  float d1 = 0.0f, d2 = 0.0f;
#pragma unroll
  for (int j = 0; j < 10; ++j) {
    int d = lane + 32 * j;
    if (d < Dd) {
      d1 += ps[d] * tk[d];
      d2 += pt[d] * sk[d];
    }
  }
  d1 = wave_sum32(d1);
  d2 = wave_sum32(d2);
  if (lane == 0) {
    out[b * Kk + k] = 2.0f * d1 - rt[i_ps] - rs[i_tk];
    out[Bb * Kk + b * Kk + k] = 2.0f * d2 - rs[i_pt] - rt[i_sk];
  }
}

// ---------------------------------------------------------------------------
extern "C" void kernel_launch(void* const* d_in, const int* in_sizes, int n_in,
                              void* d_out, int out_size, void* d_ws, size_t ws_size,
                              hipStream_t stream) {
  (void)in_sizes; (void)n_in; (void)out_size; (void)ws_size;
  const float* feat_src = (const float*)d_in[0];
  const float* feat_tgt = (const float*)d_in[1];
  const int* srcs_index = (const int*)d_in[2];
  const int* tgts_index = (const int*)d_in[3];
  const float* src_vs = (const float*)d_in[4];
  const float* tgt_vs = (const float*)d_in[5];

  float* ws = (float*)d_ws;
  float* src_hn = ws;                                  // N*D
  float* tgt_hn = ws + (size_t)Nn * Dd;                // N*D
  float* rt = ws + 2 * (size_t)Nn * Dd;                // N
  float* rs = rt + Nn;                                 // N

  // Towers + normalize: 2*N rows, 1 wave/row, 8 waves/block.
  tower_kernel<<<(2 * Nn) / 8, 256, 0, stream>>>(feat_src, feat_tgt, src_vs,
                                                 tgt_vs, src_hn, tgt_hn);
  // rt: row top-10 means of src_hn @ tgt_hn^T
  csls_topk_kernel<<<Nn / 64, 128, 0, stream>>>(src_hn, tgt_hn, rt);
  // rs: row top-10 means of tgt_hn @ src_hn^T  (columns of the first matrix)
  csls_topk_kernel<<<Nn / 64, 128, 0, stream>>>(tgt_hn, src_hn, rs);
  // Final logits: one wave per (b,k).
  logits_kernel<<<(Bb * Kk) / 8, 256, 0, stream>>>(src_hn, tgt_hn, srcs_index,
                                                   tgts_index, rt, rs,
                                                   (float*)d_out);
}